// SmoothnessRegularization_54219667145401
// MI455X (gfx1250) — compile-verified
//
#include <hip/hip_runtime.h>

#define N_PTS 16384
#define K_NB 16

typedef __attribute__((ext_vector_type(2))) float v2f;
typedef __attribute__((ext_vector_type(8))) float v8f;

// 64 blocks x 256 threads; each wave (32 lanes) owns 32 query rows.
__global__ __launch_bounds__(256) void SmoothnessReg_knn_kernel(
    const float* __restrict__ pos, const float* __restrict__ wts,
    const float* __restrict__ ls, float* __restrict__ partial)
{
    __shared__ __align__(16) float tile[8][32][16];  // per-wave 32x16 d^2 tile
    __shared__ float rowSq[256];
    __shared__ float red[256];

    const int tid   = threadIdx.x;
    const int wave  = tid >> 5;
    const int lane  = tid & 31;
    const int hi    = lane >> 4;   // lane group (0: lanes 0-15, 1: lanes 16-31)
    const int nn    = lane & 15;   // tile column (B/C/D "N"), also A row m
    const int rBase = blockIdx.x * 256 + wave * 32;
    const int gi    = rBase + lane;   // this lane's query point

    // row squared norms for the whole block
    {
        float px = pos[3*gi+0], py = pos[3*gi+1], pz = pos[3*gi+2];
        rowSq[tid] = px*px + py*py + pz*pz;
    }
    __syncthreads();

    // Loop-invariant A matrices, pre-scaled by -2 so WMMA emits d^2 directly.
    // 32-bit A 16x4 layout: lane holds row m=lane%16; VGPR0 = K0 (lanes<16)/K2 (lanes>=16),
    // VGPR1 = K1/K3.  K indexes (x,y,z,0).
    v2f a1, a2;
    {
        int m1 = rBase + nn, m2 = rBase + 16 + nn;
        float x1 = pos[3*m1+0], y1 = pos[3*m1+1], z1 = pos[3*m1+2];
        float x2 = pos[3*m2+0], y2 = pos[3*m2+1], z2 = pos[3*m2+2];
        a1.x = -2.0f * (hi ? z1 : x1);
        a1.y = hi ? 0.0f : (-2.0f * y1);
        a2.x = -2.0f * (hi ? z2 : x2);
        a2.y = hi ? 0.0f : (-2.0f * y2);
    }
    // Loop-invariant row-norm part of C: C/D layout row m = v + 8*hi
    float sqA1[8], sqA2[8];
    #pragma unroll
    for (int v = 0; v < 8; ++v) {
        sqA1[v] = rowSq[wave*32 + v + 8*hi];
        sqA2[v] = rowSq[wave*32 + 16 + v + 8*hi];
    }

    // private top-16 by squared distance
    float nd[16]; int ni[16];
    #pragma unroll
    for (int t = 0; t < 16; ++t) { nd[t] = 3.0e38f; ni[t] = 0; }
    float curMax = 3.0e38f; int curMaxPos = 0;

    for (int j0 = 0; j0 < N_PTS; j0 += 16) {
        // B (4x16): lane holds col n=lane%16; VGPR0 = K0/K2, VGPR1 = K1/K3
        int jc = j0 + nn;
        float bx = pos[3*jc+0], by = pos[3*jc+1], bz = pos[3*jc+2];
        float sqC = bx*bx + by*by + bz*bz;
        v2f b;
        b.x = hi ? bz : bx;
        b.y = hi ? 0.0f : by;

        v8f c1, c2;
        #pragma unroll
        for (int v = 0; v < 8; ++v) { c1[v] = sqA1[v] + sqC; c2[v] = sqA2[v] + sqC; }

        // d^2 tile = (-2 X_rows) x (X_cols^T) + (|x_m|^2 + |x_n|^2)
        v8f d1 = __builtin_amdgcn_wmma_f32_16x16x4_f32(false, a1, false, b,
                                                       (short)0, c1, false, false);
        v8f d2 = __builtin_amdgcn_wmma_f32_16x16x4_f32(false, a2, false, b,
                                                       (short)0, c2, false, false);

        // scatter D per documented layout: lane L, VGPR v -> row v+8*hi, col nn
        #pragma unroll
        for (int v = 0; v < 8; ++v) {
            tile[wave][v + 8*hi][nn]      = d1[v];
            tile[wave][16 + v + 8*hi][nn] = d2[v];
        }
        __syncthreads();

        // each lane scans its own row's 16 candidates (4x ds_load_b128)
        const float4* rowp = (const float4*)&tile[wave][lane][0];
        #pragma unroll
        for (int q = 0; q < 4; ++q) {
            float4 f = rowp[q];
            float dv[4] = {f.x, f.y, f.z, f.w};
            #pragma unroll
            for (int e = 0; e < 4; ++e) {
                int j = j0 + q*4 + e;
                float d2v = dv[e];
                if (j != gi && d2v < curMax) {
                    #pragma unroll
                    for (int t = 0; t < 16; ++t)
                        if (t == curMaxPos) { nd[t] = d2v; ni[t] = j; }
                    curMax = nd[0]; curMaxPos = 0;
                    #pragma unroll
                    for (int t = 1; t < 16; ++t)
                        if (nd[t] > curMax) { curMax = nd[t]; curMaxPos = t; }
                }
            }
        }
        __syncthreads();
    }

    // edge contributions: divisor max(sqrt(max(d2,1e-12)),1e-6)^2 == max(d2,1e-12)
    float wi = wts[gi];
    float s0 = ls[3*gi+0], s1 = ls[3*gi+1], s2 = ls[3*gi+2];
    float acc = 0.0f;
    #pragma unroll
    for (int t = 0; t < 16; ++t) {
        int j = ni[t];
        float nd2 = fmaxf(nd[t], 1e-12f);
        float dw = wi - wts[j];
        float e0 = s0 - ls[3*j+0];
        float e1 = s1 - ls[3*j+1];
        float e2 = s2 - ls[3*j+2];
        acc += (dw*dw + e0*e0 + e1*e1 + e2*e2) / nd2;
    }

    // deterministic block tree reduction
    red[tid] = acc;
    __syncthreads();
    for (int s = 128; s > 0; s >>= 1) {
        if (tid < s) red[tid] += red[tid + s];
        __syncthreads();
    }
    if (tid == 0) partial[blockIdx.x] = red[0];
}

__global__ void SmoothnessReg_finalize_kernel(const float* __restrict__ partial,
                                              float* __restrict__ out)
{
    __shared__ float red[64];
    int tid = threadIdx.x;
    red[tid] = partial[tid];
    __syncthreads();
    for (int s = 32; s > 0; s >>= 1) {
        if (tid < s) red[tid] += red[tid + s];
        __syncthreads();
    }
    if (tid == 0)
        out[0] = red[0] * (0.01f / (float)(N_PTS * K_NB));  // LAMBDA_S / num_edges
}

extern "C" void kernel_launch(void* const* d_in, const int* in_sizes, int n_in,
                              void* d_out, int out_size, void* d_ws, size_t ws_size,
                              hipStream_t stream)
{
    (void)in_sizes; (void)n_in; (void)out_size; (void)ws_size;
    const float* pos = (const float*)d_in[0];   // (N,3) f32
    const float* wts = (const float*)d_in[1];   // (N,)  f32
    const float* ls  = (const float*)d_in[2];   // (N,3) f32
    float* out = (float*)d_out;
    float* partial = (float*)d_ws;              // 64 floats

    const int blocks = N_PTS / 256;             // 64
    SmoothnessReg_knn_kernel<<<blocks, 256, 0, stream>>>(pos, wts, ls, partial);
    SmoothnessReg_finalize_kernel<<<1, 64, 0, stream>>>(partial, out);
}